// GCN_1layer_79027398246920
// MI455X (gfx1250) — compile-verified
//
#include <hip/hip_runtime.h>

typedef __attribute__((ext_vector_type(2))) float v2f;
typedef __attribute__((ext_vector_type(8))) float v8f;

#define N_NODES 50000
#define N_EDGES 1000000
#define IN_FEAT 256
#define HIDDEN  128
#define MTILES  (N_NODES / 16)   // 3125
#define NTILES  (HIDDEN / 16)    // 8

// ---------------------------------------------------------------------------
// Kernel 1: h = x @ W via V_WMMA_F32_16X16X4_F32 (full fp32 precision).
// Block = 256 threads = 8 waves. Block owns one 16-row M tile of x (staged in
// LDS, contiguous 16KB), wave w owns N tile w (16 columns of W). 64 chained
// WMMAs over K.
// ---------------------------------------------------------------------------
__global__ __launch_bounds__(256) void gcn_gemm(const float* __restrict__ x,
                                                const float* __restrict__ W,
                                                float* __restrict__ h) {
    __shared__ float As[16 * IN_FEAT];   // 16 KB: rows m=0..15, k=0..255
    const int mtile = blockIdx.x;
    const int tid   = threadIdx.x;

    // Stage the contiguous 16x256 fp32 tile of x with float4 loads.
    const float4* src = (const float4*)(x + (size_t)mtile * 16 * IN_FEAT);
    float4*       dst = (float4*)As;
#pragma unroll
    for (int i = 0; i < 4; ++i)
        dst[tid + 256 * i] = src[tid + 256 * i];
    __syncthreads();

    const int wave = tid >> 5;        // this wave's N tile (0..7)
    const int lane = tid & 31;
    const int half = lane >> 4;       // 0: K pair {0,1}; 1: K pair {2,3}
    const int mn   = lane & 15;       // A: row M; B: col N
    const int ncol = wave * 16 + mn;  // global column in W / h

    v8f acc = {};
#pragma unroll 8
    for (int k0 = 0; k0 < IN_FEAT; k0 += 4) {
        const int ka = k0 + 2 * half;
        // A 16x4: lane(mn,half) holds x[m=mn][ka], x[mn][ka+1]  (8B aligned)
        v2f a = *(const v2f*)(As + mn * IN_FEAT + ka);
        // B 4x16: lane(mn,half) holds W[ka][ncol], W[ka+1][ncol]
        v2f b;
        b.x = W[(size_t)ka * HIDDEN + ncol];
        b.y = W[(size_t)(ka + 1) * HIDDEN + ncol];
        acc = __builtin_amdgcn_wmma_f32_16x16x4_f32(
            /*neg_a=*/false, a, /*neg_b=*/false, b,
            /*c_mod=*/(short)0, acc, /*reuse_a=*/false, /*reuse_b=*/false);
    }

    // D 16x16 f32: VGPR r -> M = r + 8*half, N = mn.
    float* outp = h + ((size_t)mtile * 16 + 8 * half) * HIDDEN + ncol;
#pragma unroll
    for (int r = 0; r < 8; ++r)
        outp[(size_t)r * HIDDEN] = acc[r];
}

// ---------------------------------------------------------------------------
// Degree / normalization (deg counts col targets + self loop).
// ---------------------------------------------------------------------------
__global__ void gcn_init_deg(float* __restrict__ deg) {
    int i = blockIdx.x * 256 + threadIdx.x;
    if (i < N_NODES) deg[i] = 1.0f;   // self-loop contribution
}

__global__ void gcn_count_deg(const long long* __restrict__ ei,
                              float* __restrict__ deg) {
    int e = blockIdx.x * 256 + threadIdx.x;
    if (e < N_EDGES) {
        int col = (int)ei[(size_t)N_EDGES + e];   // target index
        atomicAdd(&deg[col], 1.0f);               // no-return f32 atomic
    }
}

__global__ void gcn_deg_to_dinv(float* __restrict__ deg) {
    int i = blockIdx.x * 256 + threadIdx.x;
    if (i < N_NODES) deg[i] = rsqrtf(deg[i]);     // deg >= 1 always
}

// ---------------------------------------------------------------------------
// out = b + dinv[n]^2 * h[n]   (bias + self-loop message, fully initializes out)
// ---------------------------------------------------------------------------
__global__ void gcn_self_bias(const float* __restrict__ h,
                              const float* __restrict__ dinv,
                              const float* __restrict__ b,
                              float* __restrict__ out) {
    int idx = blockIdx.x * 256 + threadIdx.x;     // over N*HIDDEN = 6.4M
    if (idx < N_NODES * HIDDEN) {
        int n = idx >> 7;
        int f = idx & (HIDDEN - 1);
        float di = dinv[n];
        out[idx] = b[f] + di * di * h[idx];
    }
}

// ---------------------------------------------------------------------------
// Edge scatter: one wave per edge, lane handles 4 features (float4 gather,
// 4 no-return global f32 atomic adds). h/out both L2-resident (25.6MB each).
// ---------------------------------------------------------------------------
__global__ __launch_bounds__(256) void gcn_edge_scatter(
        const long long* __restrict__ ei,
        const float* __restrict__ h,
        const float* __restrict__ dinv,
        float* __restrict__ out) {
    int e    = blockIdx.x * 8 + (threadIdx.x >> 5);
    int lane = threadIdx.x & 31;
    if (e < N_EDGES) {
        int row = (int)ei[e];                        // source (gather)
        int col = (int)ei[(size_t)N_EDGES + e];      // target (scatter)
        float norm = dinv[row] * dinv[col];
        const float4 hv = *(const float4*)(h + (size_t)row * HIDDEN + lane * 4);
        float* op = out + (size_t)col * HIDDEN + lane * 4;
        atomicAdd(op + 0, norm * hv.x);
        atomicAdd(op + 1, norm * hv.y);
        atomicAdd(op + 2, norm * hv.z);
        atomicAdd(op + 3, norm * hv.w);
    }
}

// ---------------------------------------------------------------------------
extern "C" void kernel_launch(void* const* d_in, const int* in_sizes, int n_in,
                              void* d_out, int out_size, void* d_ws, size_t ws_size,
                              hipStream_t stream) {
    const float*     x  = (const float*)d_in[0];      // [50000,256] f32
    const long long* ei = (const long long*)d_in[1];  // [2,1000000] i64
    const float*     W  = (const float*)d_in[2];      // [256,128] f32
    const float*     b  = (const float*)d_in[3];      // [128] f32
    float*           out = (float*)d_out;             // [50000,128] f32

    float* h   = (float*)d_ws;                        // 25.6 MB
    float* deg = h + (size_t)N_NODES * HIDDEN;        // 200 KB (deg -> dinv in place)

    // 1) h = x @ W  (WMMA fp32)
    gcn_gemm<<<MTILES, 256, 0, stream>>>(x, W, h);

    // 2) degree (with self-loops) -> dinv
    gcn_init_deg<<<(N_NODES + 255) / 256, 256, 0, stream>>>(deg);
    gcn_count_deg<<<(N_EDGES + 255) / 256, 256, 0, stream>>>(ei, deg);
    gcn_deg_to_dinv<<<(N_NODES + 255) / 256, 256, 0, stream>>>(deg);

    // 3) out = b + dinv^2 * h (initializes every element of d_out)
    gcn_self_bias<<<(N_NODES * HIDDEN + 255) / 256, 256, 0, stream>>>(h, deg, b, out);

    // 4) edge messages: out[col] += dinv[row]*dinv[col]*h[row]
    gcn_edge_scatter<<<N_EDGES / 8, 256, 0, stream>>>(ei, h, deg, out);
}